// SRenderY_51067161149879
// MI455X (gfx1250) — compile-verified
//
#include <hip/hip_runtime.h>
#include <hip/hip_bf16.h>

typedef float v2f __attribute__((ext_vector_type(2)));
typedef float v8f __attribute__((ext_vector_type(8)));

#define B_ 4
#define V_ 5023
#define F_ 9976
#define H_ 512
#define W_ 512
#define U_ 256
#define HW_ (H_ * W_)

// SH constant factors (CONST_FACTOR), padded to 12 for K=4 WMMA chunks
__constant__ float c_CF[12] = {
    0.28209479177387814f,
    1.0233267079464885f, 1.0233267079464885f, 1.0233267079464885f,
    0.8580855308302045f, 0.8580855308302045f, 0.8580855308302045f,
    0.4290427654151022f,
    0.24770795610037571f,
    0.0f, 0.0f, 0.0f
};

// ---------------- precompute: vertex normals ----------------

__global__ __launch_bounds__(256) void zero_kernel(float* __restrict__ p, int n) {
    int i = blockIdx.x * 256 + threadIdx.x;
    if (i < n) p[i] = 0.0f;
}

__global__ __launch_bounds__(256) void scatter_normals_kernel(
    const float* __restrict__ verts, const int* __restrict__ faces,
    float* __restrict__ nacc)
{
    int t = blockIdx.x * 256 + threadIdx.x;
    if (t >= B_ * F_) return;
    int b = t / F_;
    int f = t - b * F_;
    int i0 = faces[f * 3 + 0];
    int i1 = faces[f * 3 + 1];
    int i2 = faces[f * 3 + 2];
    const float* vb = verts + (size_t)b * V_ * 3;
    float p0x = vb[i0 * 3 + 0], p0y = vb[i0 * 3 + 1], p0z = vb[i0 * 3 + 2];
    float p1x = vb[i1 * 3 + 0], p1y = vb[i1 * 3 + 1], p1z = vb[i1 * 3 + 2];
    float p2x = vb[i2 * 3 + 0], p2y = vb[i2 * 3 + 1], p2z = vb[i2 * 3 + 2];
    float* nb = nacc + (size_t)b * V_ * 3;

    // n[i1] += cross(p2-p1, p0-p1)
    {
        float ax = p2x - p1x, ay = p2y - p1y, az = p2z - p1z;
        float bx = p0x - p1x, by = p0y - p1y, bz = p0z - p1z;
        atomicAdd(&nb[i1 * 3 + 0], ay * bz - az * by);
        atomicAdd(&nb[i1 * 3 + 1], az * bx - ax * bz);
        atomicAdd(&nb[i1 * 3 + 2], ax * by - ay * bx);
    }
    // n[i2] += cross(p0-p2, p1-p2)
    {
        float ax = p0x - p2x, ay = p0y - p2y, az = p0z - p2z;
        float bx = p1x - p2x, by = p1y - p2y, bz = p1z - p2z;
        atomicAdd(&nb[i2 * 3 + 0], ay * bz - az * by);
        atomicAdd(&nb[i2 * 3 + 1], az * bx - ax * bz);
        atomicAdd(&nb[i2 * 3 + 2], ax * by - ay * bx);
    }
    // n[i0] += cross(p1-p0, p2-p0)
    {
        float ax = p1x - p0x, ay = p1y - p0y, az = p1z - p0z;
        float bx = p2x - p0x, by = p2y - p0y, bz = p2z - p0z;
        atomicAdd(&nb[i0 * 3 + 0], ay * bz - az * by);
        atomicAdd(&nb[i0 * 3 + 1], az * bx - ax * bz);
        atomicAdd(&nb[i0 * 3 + 2], ax * by - ay * bx);
    }
}

__global__ __launch_bounds__(256) void normalize_kernel(float* __restrict__ n) {
    int t = blockIdx.x * 256 + threadIdx.x;
    if (t >= B_ * V_) return;
    float x = n[t * 3 + 0], y = n[t * 3 + 1], z = n[t * 3 + 2];
    float d = fmaxf(sqrtf(x * x + y * y + z * z), 1e-6f);
    float inv = 1.0f / d;
    n[t * 3 + 0] = x * inv;
    n[t * 3 + 1] = y * inv;
    n[t * 3 + 2] = z * inv;
}

// ---------------- fused per-pixel render ----------------
// One thread per pixel. Per wave (32 px): SH shading via V_WMMA_F32_16X16X4_F32:
//   shading(16px x 3col) = SHbasis(16x12) x (cf*lights)(12x3), 3 K=4 chunks.

__global__ __launch_bounds__(256) void render_kernel(
    const float* __restrict__ bary,        // (B,H,W,1,3)
    const int*   __restrict__ p2f,         // (B,H,W,1)
    const int*   __restrict__ faces,       // (F,3)
    const float* __restrict__ uvc,         // (1,F,3,3)
    const float* __restrict__ nrm,         // (B,V,3) normalized
    const float* __restrict__ albedos,     // (B,3,U,U)
    const float* __restrict__ lights,      // (B,9,3)
    float*       __restrict__ out)         // (B,3,H,W)
{
    __shared__ float cfl[12][16];      // B operand staging: cf[j]*lights[b][j][c]
    __shared__ float wsh[8][512];      // per-wave: SH basis tile, then D dump

    int tid = blockIdx.x * 256 + threadIdx.x;
    int b   = tid >> 18;               // H*W = 2^18, so b uniform per block
    int rem = tid & (HW_ - 1);
    int h   = rem >> 9;                // W = 512
    int w   = rem & (W_ - 1);

    // stage B-operand values once per block
    if (threadIdx.x < 192) {
        int j = threadIdx.x >> 4;
        int c = threadIdx.x & 15;
        float v = 0.0f;
        if (j < 9 && c < 3) v = c_CF[j] * lights[(b * 9 + j) * 3 + c];
        cfl[j][c] = v;
    }

    // ---- per-pixel gathers + barycentric interpolation ----
    int fi = p2f[tid];
    bool valid = fi >= 0;
    int fs = valid ? fi : 0;
    float b0 = bary[(size_t)tid * 3 + 0];
    float b1 = bary[(size_t)tid * 3 + 1];
    float b2 = bary[(size_t)tid * 3 + 2];

    int v0 = faces[fs * 3 + 0];
    int v1 = faces[fs * 3 + 1];
    int v2 = faces[fs * 3 + 2];

    const float* uf = uvc + (size_t)fs * 9;   // (3 verts x 3 comps)
    float u  = b0 * uf[0] + b1 * uf[3] + b2 * uf[6];
    float vv = b0 * uf[1] + b1 * uf[4] + b2 * uf[7];

    const float* nb = nrm + (size_t)b * V_ * 3;
    float nx = b0 * nb[v0 * 3 + 0] + b1 * nb[v1 * 3 + 0] + b2 * nb[v2 * 3 + 0];
    float ny = b0 * nb[v0 * 3 + 1] + b1 * nb[v1 * 3 + 1] + b2 * nb[v2 * 3 + 1];
    float nz = b0 * nb[v0 * 3 + 2] + b1 * nb[v1 * 3 + 2] + b2 * nb[v2 * 3 + 2];

    if (!valid) { u = 0.0f; vv = 0.0f; nx = 0.0f; ny = 0.0f; nz = 0.0f; }

    // ---- bilinear albedo sample (grid_sample, zero-pad border) ----
    float x = (u + 1.0f) * (U_ * 0.5f) - 0.5f;
    float y = (vv + 1.0f) * (U_ * 0.5f) - 0.5f;
    float x0f = floorf(x), y0f = floorf(y);
    float wx = x - x0f, wy = y - y0f;
    int x0 = (int)x0f, y0 = (int)y0f;

    const float* ab = albedos + (size_t)b * 3 * U_ * U_;
    float alb[3] = {0.0f, 0.0f, 0.0f};
#pragma unroll
    for (int dy = 0; dy < 2; ++dy) {
#pragma unroll
        for (int dx = 0; dx < 2; ++dx) {
            int xi = x0 + dx, yi = y0 + dy;
            bool ok = (xi >= 0) & (xi < U_) & (yi >= 0) & (yi < U_);
            int xc = min(max(xi, 0), U_ - 1);
            int yc = min(max(yi, 0), U_ - 1);
            float wgt = (dx ? wx : (1.0f - wx)) * (dy ? wy : (1.0f - wy));
            wgt = ok ? wgt : 0.0f;
            size_t base = (size_t)yc * U_ + xc;
#pragma unroll
            for (int c = 0; c < 3; ++c)
                alb[c] += wgt * ab[(size_t)c * U_ * U_ + base];
        }
    }

    // ---- SH basis -> LDS staging tile ----
    int wave = threadIdx.x >> 5;
    int lane = threadIdx.x & 31;
    float* S = &wsh[wave][0];          // [pixel 0..31][16]
    S[lane * 16 + 0]  = 1.0f;
    S[lane * 16 + 1]  = nx;
    S[lane * 16 + 2]  = ny;
    S[lane * 16 + 3]  = nz;
    S[lane * 16 + 4]  = nx * ny;
    S[lane * 16 + 5]  = nx * nz;
    S[lane * 16 + 6]  = ny * nz;
    S[lane * 16 + 7]  = nx * nx - ny * ny;
    S[lane * 16 + 8]  = 3.0f * nz * nz - 1.0f;
    S[lane * 16 + 9]  = 0.0f;
    S[lane * 16 + 10] = 0.0f;
    S[lane * 16 + 11] = 0.0f;
    __syncthreads();

    // ---- WMMA: D = A(16x4 f32) x B(4x16 f32) + C, accumulate 3 K-chunks ----
    // A layout (ISA 16x4 f32): lanes 0-15 rows M, VGPR0 = {K0 | K2}, VGPR1 = {K1 | K3}
    int row = lane & 15;
    bool hi = lane >= 16;
    int ke = hi ? 2 : 0;   // K offset for .x
    int ko = hi ? 3 : 1;   // K offset for .y

    v8f acc0 = {};  // pixels 0..15 of this wave
    v8f acc1 = {};  // pixels 16..31
#pragma unroll
    for (int kb = 0; kb < 12; kb += 4) {
        v2f a0, a1, bb;
        a0.x = S[row * 16 + kb + ke];
        a0.y = S[row * 16 + kb + ko];
        a1.x = S[(16 + row) * 16 + kb + ke];
        a1.y = S[(16 + row) * 16 + kb + ko];
        bb.x = cfl[kb + ke][row];
        bb.y = cfl[kb + ko][row];
        acc0 = __builtin_amdgcn_wmma_f32_16x16x4_f32(false, a0, false, bb,
                                                     (short)0, acc0, false, false);
        acc1 = __builtin_amdgcn_wmma_f32_16x16x4_f32(false, a1, false, bb,
                                                     (short)0, acc1, false, false);
    }
    __syncthreads();

    // ---- dump D through LDS (D layout: VGPR r, lane l -> (M=r+8*(l>=16), N=l&15)) ----
    {
        int mofs = hi ? 8 : 0;
        int col = lane & 15;
#pragma unroll
        for (int r = 0; r < 8; ++r) {
            S[(r + mofs) * 16 + col]      = acc0[r];
            S[(16 + r + mofs) * 16 + col] = acc1[r];
        }
    }
    __syncthreads();

    float s0 = S[lane * 16 + 0];
    float s1 = S[lane * 16 + 1];
    float s2 = S[lane * 16 + 2];

    size_t obase = ((size_t)b * 3 * H_ + h) * W_ + w;
    out[obase]                        = alb[0] * s0;
    out[obase + (size_t)HW_]          = alb[1] * s1;
    out[obase + (size_t)(2 * HW_)]    = alb[2] * s2;
}

// ---------------- host launch ----------------

extern "C" void kernel_launch(void* const* d_in, const int* in_sizes, int n_in,
                              void* d_out, int out_size, void* d_ws, size_t ws_size,
                              hipStream_t stream) {
    const float* vertices = (const float*)d_in[0];
    // d_in[1] transformed_vertices: dead w.r.t. the output (only fed t_normals,
    // which never reach `images`) — intentionally unused.
    const float* albedos  = (const float*)d_in[2];
    const float* lights   = (const float*)d_in[3];
    const float* uvc      = (const float*)d_in[4];
    const float* bary     = (const float*)d_in[5];
    const int*   faces    = (const int*)d_in[6];
    const int*   p2f      = (const int*)d_in[7];
    float* out = (float*)d_out;

    float* nacc = (float*)d_ws;            // (B,V,3) accumulator, normalized in place
    (void)in_sizes; (void)n_in; (void)out_size; (void)ws_size;

    int nN = B_ * V_ * 3;
    zero_kernel<<<(nN + 255) / 256, 256, 0, stream>>>(nacc, nN);
    scatter_normals_kernel<<<(B_ * F_ + 255) / 256, 256, 0, stream>>>(vertices, faces, nacc);
    normalize_kernel<<<(B_ * V_ + 255) / 256, 256, 0, stream>>>(nacc);
    render_kernel<<<(B_ * HW_) / 256, 256, 0, stream>>>(
        bary, p2f, faces, uvc, nacc, albedos, lights, out);
}